// GraphAttentionLayer_4879082848765
// MI455X (gfx1250) — compile-verified
//
#include <hip/hip_runtime.h>
#include <hip/hip_bf16.h>
#include <math.h>

// ---------------------------------------------------------------------------
// GAT layer for MI455X (gfx1250, wave32).
// GEMM uses V_WMMA_F32_16X16X4_F32 (exact fp32 — kernel is bandwidth-bound,
// so no reason to downcast). Edge phase is gather + global softmax + atomic
// scatter-add, which dominates runtime (~410MB of L2-resident gathers + 102M
// fp32 atomics vs a 1.64 GFLOP GEMM).
// ---------------------------------------------------------------------------

typedef __attribute__((ext_vector_type(2))) float v2f;
typedef __attribute__((ext_vector_type(8))) float v8f;

#define N_NODES 50000
#define N_EDGES 800000
#define D       128
#define M_TILES (N_NODES / 16)   // 3125, exact
#define ALPHA   0.2f

// ---- ordered-float encoding for atomicMax on uint ----
__device__ __forceinline__ unsigned enc_f32(float f) {
    unsigned u = __float_as_uint(f);
    return (u & 0x80000000u) ? ~u : (u | 0x80000000u);
}
__device__ __forceinline__ float dec_f32(unsigned u) {
    return (u & 0x80000000u) ? __uint_as_float(u & 0x7FFFFFFFu)
                             : __uint_as_float(~u);
}
#define ENC_NEG_INF 0x007FFFFFu   // enc_f32(-inf)

// ---------------------------------------------------------------------------
// 0) init: zero output accumulator, reset softmax max/sum (every call --
//    harness graph-replays without re-poisoning, so this must be per-launch).
// ---------------------------------------------------------------------------
__global__ void gat_init(float* __restrict__ out, unsigned* __restrict__ red) {
    const int stride = gridDim.x * blockDim.x;
    const int n = N_NODES * D;
    for (int i = blockIdx.x * blockDim.x + threadIdx.x; i < n; i += stride)
        out[i] = 0.0f;
    if (blockIdx.x == 0 && threadIdx.x == 0) {
        red[0] = ENC_NEG_INF;            // running max (ordered encoding)
        ((float*)red)[1] = 0.0f;         // exp-sum
    }
}

// ---------------------------------------------------------------------------
// 1) h = x @ W via V_WMMA_F32_16X16X4_F32.
//    One wave computes a 16x128 row block: 8 C-tiles, K-loop of 32 WMMAs each.
//    fp32 operand layouts (ISA 7.12.2):
//      A 16x4 : lane = row (lane&15); VGPR0/1 = K = 2*(lane>>4) + {0,1}
//      B 4x16 : lane = col (lane&15); VGPR v  = K = 2*(lane>>4) + v
//      C 16x16: VGPR r -> row r + 8*(lane>>4), col = lane&15
// ---------------------------------------------------------------------------
__global__ void gat_gemm(const float* __restrict__ x,
                         const float* __restrict__ W,
                         float* __restrict__ h) {
    const int wave  = threadIdx.x >> 5;
    const int mtile = blockIdx.x * (blockDim.x >> 5) + wave;
    if (mtile >= M_TILES) return;                 // wave-uniform: EXEC stays all-1s

    const int lane  = threadIdx.x & 31;
    const int mrow  = lane & 15;                  // A row / B col / C col
    const int khalf = lane >> 4;                  // selects K pair

    const float* __restrict__ arow = x + (mtile * 16 + mrow) * D;

    v8f acc[8] = {};                              // 8 N-tiles of 16 cols

    for (int k = 0; k < D; k += 4) {
        const int ka = k + 2 * khalf;
        v2f a;
        a.x = arow[ka];
        a.y = arow[ka + 1];
        const float* __restrict__ w0 = W + ka * D + mrow;
#pragma unroll
        for (int t = 0; t < 8; ++t) {
            v2f b;
            b.x = w0[t * 16];                     // K = ka,   N = t*16+mrow
            b.y = w0[D + t * 16];                 // K = ka+1
            acc[t] = __builtin_amdgcn_wmma_f32_16x16x4_f32(
                false, a, false, b, (short)0, acc[t], false, false);
        }
    }

    float* __restrict__ hrow = h + (mtile * 16) * D;
#pragma unroll
    for (int t = 0; t < 8; ++t)
#pragma unroll
        for (int r = 0; r < 8; ++r)
            hrow[(r + 8 * khalf) * D + t * 16 + mrow] = acc[t][r];
}

// ---------------------------------------------------------------------------
// 2) s1[i] = h[i,:].a1 ; s2[i] = h[i,:].a2  -- one wave per node.
// ---------------------------------------------------------------------------
__global__ void gat_scores(const float* __restrict__ h,
                           const float* __restrict__ a,
                           float* __restrict__ s1,
                           float* __restrict__ s2) {
    const int node = (blockIdx.x * blockDim.x + threadIdx.x) >> 5;
    if (node >= N_NODES) return;
    const int lane = threadIdx.x & 31;
    const float* __restrict__ row = h + node * D;

    float p1 = 0.0f, p2 = 0.0f;
#pragma unroll
    for (int j = 0; j < 4; ++j) {
        const int c = lane + 32 * j;
        const float v = row[c];
        p1 += v * a[c];
        p2 += v * a[D + c];
    }
#pragma unroll
    for (int off = 16; off > 0; off >>= 1) {
        p1 += __shfl_down(p1, off, 32);
        p2 += __shfl_down(p2, off, 32);
    }
    if (lane == 0) { s1[node] = p1; s2[node] = p2; }
}

// ---------------------------------------------------------------------------
// 3) per-edge leaky-relu score + global max (stable softmax pass 1).
// ---------------------------------------------------------------------------
__global__ void gat_edge_score(const float* __restrict__ s1,
                               const float* __restrict__ s2,
                               const int* __restrict__ ei,
                               float* __restrict__ esc,
                               unsigned* __restrict__ red) {
    __shared__ float smax[256];
    const int e = blockIdx.x * 256 + threadIdx.x;
    float v = -INFINITY;
    if (e < N_EDGES) {
        const int s = ei[e];
        const int d = ei[N_EDGES + e];
        v = s1[s] + s2[d];
        v = (v > 0.0f) ? v : ALPHA * v;
        esc[e] = v;
    }
    smax[threadIdx.x] = v;
    __syncthreads();
#pragma unroll
    for (int off = 128; off > 0; off >>= 1) {
        if (threadIdx.x < off)
            smax[threadIdx.x] = fmaxf(smax[threadIdx.x], smax[threadIdx.x + off]);
        __syncthreads();
    }
    if (threadIdx.x == 0) atomicMax(red, enc_f32(smax[0]));
}

// ---------------------------------------------------------------------------
// 4) exp(e - max) in place + global sum (stable softmax pass 2).
// ---------------------------------------------------------------------------
__global__ void gat_edge_exp(float* __restrict__ esc,
                             unsigned* __restrict__ red) {
    __shared__ float ssum[256];
    const float maxv = dec_f32(red[0]);
    const int e = blockIdx.x * 256 + threadIdx.x;
    float v = 0.0f;
    if (e < N_EDGES) {
        v = __expf(esc[e] - maxv);
        esc[e] = v;
    }
    ssum[threadIdx.x] = v;
    __syncthreads();
#pragma unroll
    for (int off = 128; off > 0; off >>= 1) {
        if (threadIdx.x < off)
            ssum[threadIdx.x] += ssum[threadIdx.x + off];
        __syncthreads();
    }
    if (threadIdx.x == 0) atomicAdd(((float*)red) + 1, ssum[0]);
}

// ---------------------------------------------------------------------------
// 5) out[dst] += (esc[e]/sum) * h[src]  -- one wave per edge, lanes cover D.
// ---------------------------------------------------------------------------
__global__ void gat_aggregate(const float* __restrict__ h,
                              const float* __restrict__ esc,
                              const int* __restrict__ ei,
                              const unsigned* __restrict__ red,
                              float* __restrict__ out) {
    const int e = (blockIdx.x * blockDim.x + threadIdx.x) >> 5;
    if (e >= N_EDGES) return;
    const int lane = threadIdx.x & 31;

    const float inv = 1.0f / ((const float*)red)[1];
    const int s = ei[e];
    const int d = ei[N_EDGES + e];
    const float att = esc[e] * inv;

    const float* __restrict__ hs = h + s * D;
    float* __restrict__ od = out + d * D;
#pragma unroll
    for (int j = 0; j < 4; ++j) {
        const int c = lane + 32 * j;
        atomicAdd(&od[c], att * hs[c]);
    }
}

// ---------------------------------------------------------------------------
extern "C" void kernel_launch(void* const* d_in, const int* in_sizes, int n_in,
                              void* d_out, int out_size, void* d_ws, size_t ws_size,
                              hipStream_t stream) {
    (void)in_sizes; (void)n_in; (void)out_size; (void)ws_size;

    const float* x  = (const float*)d_in[0];   // [N_NODES, D]
    const float* W  = (const float*)d_in[1];   // [D, D]
    const float* a  = (const float*)d_in[2];   // [2*D]
    const int*   ei = (const int*)d_in[3];     // [2, N_EDGES]
    float* out = (float*)d_out;                // [N_NODES, D]

    // workspace layout
    float*    h   = (float*)d_ws;              // N_NODES*D
    float*    s1  = h + (size_t)N_NODES * D;   // N_NODES
    float*    s2  = s1 + N_NODES;              // N_NODES
    float*    esc = s2 + N_NODES;              // N_EDGES
    unsigned* red = (unsigned*)(esc + N_EDGES);// [0]=max(enc), [1]=sum

    // 0) re-init accumulators every call (graph replay safe)
    gat_init<<<2048, 256, 0, stream>>>(out, red);

    // 1) h = x @ W  (WMMA fp32; 8 waves/block, 1 M-tile per wave)
    {
        const int waves_per_block = 8;
        const int blocks = (M_TILES + waves_per_block - 1) / waves_per_block;
        gat_gemm<<<blocks, waves_per_block * 32, 0, stream>>>(x, W, h);
    }

    // 2) per-node attention scalars (1 wave per node)
    gat_scores<<<(N_NODES * 32 + 255) / 256, 256, 0, stream>>>(h, a, s1, s2);

    // 3) edge scores + global max
    gat_edge_score<<<(N_EDGES + 255) / 256, 256, 0, stream>>>(s1, s2, ei, esc, red);

    // 4) exp + global sum
    gat_edge_exp<<<(N_EDGES + 255) / 256, 256, 0, stream>>>(esc, red);

    // 5) weighted scatter-add (1 wave per edge)
    gat_aggregate<<<(N_EDGES * 32 + 255) / 256, 256, 0, stream>>>(h, esc, ei, red, out);
}